// RCell_70171175682437
// MI455X (gfx1250) — compile-verified
//
#include <hip/hip_runtime.h>
#include <stdint.h>

// ---------------------------------------------------------------------------
// MI455X (gfx1250) implementation of the 2x2 stochastic-filter scan.
//
// 2048 independent sequential chains -> 1 lane per chain, 32 lanes (1 wave)
// per workgroup, 64 workgroups. Inputs streamed with the Tensor Data Mover
// (tensor_load_to_lds, TENSORcnt, double buffered, TDM LDS padding for
// bank-conflict-free reads). Outputs staged in LDS and written back with
// coalesced non-temporal b64 stores.
// ---------------------------------------------------------------------------

typedef unsigned int u32;
typedef u32   __attribute__((ext_vector_type(4))) u32x4;
typedef int   __attribute__((ext_vector_type(8))) i32x8;
typedef int   __attribute__((ext_vector_type(4))) i32x4;
typedef float __attribute__((ext_vector_type(2))) f32x2;

#define ROWS      32                 // batch rows per workgroup == lanes per wave
#define T_TOTAL   4096
#define TS        128                // time steps per tile
#define NTILES    (T_TOTAL / TS)
#define IN_ROW    (3 * TS)           // 384 dwords per input row (t,dy0,dy1)*TS
#define IN_PITCH  (IN_ROW + 3)       // 387: TDM pads 1 dword per 128 -> 3 mod 64 banks
#define OUT_PITCH (2 * TS + 2)       // 258: 2 mod 64 banks, conflict-free both phases

// Issue one TDM 2D tile load: 32 rows x 384 floats, global row stride = T*3
// floats, packed into LDS with 1-dword padding every 128 dwords.
__device__ __forceinline__ void tdm_load_tile(u32 lds_off, const float* gptr) {
    unsigned long long ga = (unsigned long long)(uintptr_t)gptr;
    u32x4 g0;
    g0.x = 1u;                                            // count=1 (valid descriptor)
    g0.y = lds_off;                                       // LDS byte address
    g0.z = (u32)ga;                                       // global_addr[31:0]
    g0.w = (u32)((ga >> 32) & 0x1FFFFFFu) | (2u << 30);   // global_addr[56:32] | type=2

    i32x8 g1;
    g1[0] = (2 << 16)        // data_size = 4 bytes
          | (1 << 20)        // pad_enable
          | (6 << 22);       // pad_interval code 6 = every 128 dwords (+pad_amount code 0 = 1 dword)
    g1[1] = (int)((IN_ROW & 0xFFFF) << 16);               // tensor_dim0[15:0]
    g1[2] = (int)((IN_ROW >> 16) | (ROWS << 16));         // tensor_dim0[31:16] | tensor_dim1[15:0]
    g1[3] = (int)((ROWS >> 16) | (IN_ROW << 16));         // tensor_dim1[31:16] | tile_dim0
    g1[4] = (int)(ROWS);                                  // tile_dim1=32, tile_dim2=0
    g1[5] = (int)(T_TOTAL * 3);                           // tensor_dim0_stride (elements)
    g1[6] = 0;                                            // stride0 hi / stride1 lo
    g1[7] = 0;                                            // stride1 hi
    i32x4 z4 = {0, 0, 0, 0};                              // groups 2/3: dims 2..4 unused
    i32x8 z8 = {0, 0, 0, 0, 0, 0, 0, 0};                  // trailing operand (clang-23 form)
    __builtin_amdgcn_tensor_load_to_lds(g0, g1, z4, z4, z8, 0); // TENSORcnt++
}

__global__ __launch_bounds__(ROWS) void
rcell_kernel(const float* __restrict__ in,      // (B, T, 3)
             const float* __restrict__ A_,      // (2,2)
             const float* __restrict__ D_,      // (2,2)
             const float* __restrict__ E_,      // (2,2)
             const float* __restrict__ Bm,      // (2,2)
             const float* __restrict__ covin,   // (2,2)
             const float* __restrict__ xs,      // (2,)
             const float* __restrict__ x0p,     // (B,2)
             float* __restrict__ out)           // (B, T, 2)
{
    extern __shared__ float smem[];
    float* sin0 = smem;                                  // ROWS*IN_PITCH
    float* sin1 = smem + ROWS * IN_PITCH;
    float* sout = smem + 2 * ROWS * IN_PITCH;            // ROWS*OUT_PITCH

    const int lane = threadIdx.x;                        // 0..31 (wave32)
    const int b0   = blockIdx.x * ROWS;
    const int b    = b0 + lane;

    const float SQ2 = 1.41421356237309515f;
    const float ISQ = 0.70710678118654752f;
    const float DTc = 1e-4f;

    // Tiny uniform constants (compiler keeps these as scalar loads).
    const float a00 = A_[0], a01 = A_[1], a10 = A_[2], a11 = A_[3];
    const float b00 = Bm[0], b01 = Bm[1], b10 = Bm[2], b11 = Bm[3];
    // negBT[i][j] = -sqrt(2) * B[j][i]
    const float n00 = -SQ2 * b00, n01 = -SQ2 * b10;
    const float n10 = -SQ2 * b01, n11 = -SQ2 * b11;
    const float e00 = E_[0], e01 = E_[1], e10 = E_[2], e11 = E_[3];
    const float d00 = D_[0], d01 = D_[1], d10 = D_[2], d11 = D_[3];
    const float k0 = 10.0f * DTc * xs[0], k1 = 10.0f * DTc * xs[1];

    // Per-lane state
    float x0v = x0p[2 * b + 0], x1v = x0p[2 * b + 1];
    float c00 = covin[0], c01 = covin[1], c10 = covin[2], c11 = covin[3];

    // LDS byte offsets for the TDM descriptor (low 32 bits of the generic
    // LDS address carry the workgroup-relative LDS offset).
    const u32 base    = (u32)(uintptr_t)smem;
    const u32 in0_off = base;
    const u32 in1_off = base + (u32)(ROWS * IN_PITCH * 4);

    const float* gin = in + (size_t)b0 * T_TOTAL * 3;

    // Prologue: DMA tile 0
    tdm_load_tile(in0_off, gin);

    for (int k = 0; k < NTILES; ++k) {
        // Prefetch tile k+1 into the other buffer; wait so exactly the
        // prefetch may remain in flight while we compute on tile k.
        if (k + 1 < NTILES) {
            asm volatile("s_wait_dscnt 0x0" ::: "memory");   // prior reads of that buffer drained
            tdm_load_tile(((k + 1) & 1) ? in1_off : in0_off,
                          gin + (size_t)(k + 1) * IN_ROW);
            __builtin_amdgcn_s_wait_tensorcnt(1);            // tile k landed
        } else {
            __builtin_amdgcn_s_wait_tensorcnt(0);
        }
        asm volatile("" ::: "memory");

        const float* srow = ((k & 1) ? sin1 : sin0) + lane * IN_PITCH;
        float*       orow = sout + lane * OUT_PITCH;

        for (int s = 0; s < TS; ++s) {
            // TDM padding: linear index c maps to LDS index c + c/128
            const int c0 = 3 * s, c1 = 3 * s + 1, c2 = 3 * s + 2;
            const float tv  = srow[c0 + (c0 >> 7)];
            const float dy0 = srow[c1 + (c1 >> 7)];
            const float dy1 = srow[c2 + (c2 >> 7)];

            // XiCov = (E - cov @ B) / sqrt(2)
            const float xc00 = (e00 - (c00 * b00 + c01 * b10)) * ISQ;
            const float xc01 = (e01 - (c00 * b01 + c01 * b11)) * ISQ;
            const float xc10 = (e10 - (c10 * b00 + c11 * b10)) * ISQ;
            const float xc11 = (e11 - (c10 * b01 + c11 * b11)) * ISQ;

            // out = x @ negBT^T * DT   (uses pre-update x)
            orow[2 * s + 0] = (x0v * n00 + x1v * n01) * DTc;
            orow[2 * s + 1] = (x0v * n10 + x1v * n11) * DTc;

            // G = A - XiCov @ negBT
            const float g00 = a00 - (xc00 * n00 + xc01 * n10);
            const float g01 = a01 - (xc00 * n01 + xc01 * n11);
            const float g10 = a10 - (xc10 * n00 + xc11 * n10);
            const float g11 = a11 - (xc10 * n01 + xc11 * n11);

            // drive = 10*cos(0.2*pi*t)*DT*x_signal  (v_cos_f32 path)
            const float cv  = __cosf(0.62831853071795864f * tv);
            const float dr0 = cv * k0, dr1 = cv * k1;

            const float nx0 = x0v + (g00 * x0v + g01 * x1v) * DTc
                                  + (xc00 * dy0 + xc01 * dy1) + dr0;
            const float nx1 = x1v + (g10 * x0v + g11 * x1v) * DTc
                                  + (xc10 * dy0 + xc11 * dy1) + dr1;

            // cov += (A@cov + cov@A^T + D - 2*XiCov@XiCov^T) * DT
            const float ac00 = a00 * c00 + a01 * c10, ac01 = a00 * c01 + a01 * c11;
            const float ac10 = a10 * c00 + a11 * c10, ac11 = a10 * c01 + a11 * c11;
            const float ca00 = c00 * a00 + c01 * a01, ca01 = c00 * a10 + c01 * a11;
            const float ca10 = c10 * a00 + c11 * a01, ca11 = c10 * a10 + c11 * a11;
            const float xx00 = xc00 * xc00 + xc01 * xc01;
            const float xx01 = xc00 * xc10 + xc01 * xc11;
            const float xx10 = xc10 * xc00 + xc11 * xc01;
            const float xx11 = xc10 * xc10 + xc11 * xc11;
            c00 += (ac00 + ca00 + d00 - 2.0f * xx00) * DTc;
            c01 += (ac01 + ca01 + d01 - 2.0f * xx01) * DTc;
            c10 += (ac10 + ca10 + d10 - 2.0f * xx10) * DTc;
            c11 += (ac11 + ca11 + d11 - 2.0f * xx11) * DTc;
            x0v = nx0; x1v = nx1;
        }

        // Coalesced non-temporal writeout of the staged output tile:
        // row r -> out[(b0+r)*T*2 + k*TS*2 ...], 32 lanes x float2 = 256B/op.
        __syncthreads();
        for (int r = 0; r < ROWS; ++r) {
            const f32x2* src = (const f32x2*)(sout + r * OUT_PITCH);
            f32x2* dst = (f32x2*)(out + ((size_t)(b0 + r) * T_TOTAL + (size_t)k * TS) * 2);
#pragma unroll
            for (int i = 0; i < (TS * 2) / 64; ++i) {     // 4 iterations
                __builtin_nontemporal_store(src[i * 32 + lane], dst + i * 32 + lane);
            }
        }
        __syncthreads();   // out tile reusable next iteration
    }
}

extern "C" void kernel_launch(void* const* d_in, const int* in_sizes, int n_in,
                              void* d_out, int out_size, void* d_ws, size_t ws_size,
                              hipStream_t stream) {
    const float* in    = (const float*)d_in[0];
    const float* A_    = (const float*)d_in[1];
    const float* D_    = (const float*)d_in[2];
    const float* E_    = (const float*)d_in[3];
    const float* Bm    = (const float*)d_in[4];
    const float* covin = (const float*)d_in[5];
    const float* xs    = (const float*)d_in[6];
    const float* x0p   = (const float*)d_in[7];
    float* out = (float*)d_out;

    const int B = in_sizes[7] / 2;            // x0 is (B,2) -> B = 2048
    const size_t shmem =
        (size_t)(2 * ROWS * IN_PITCH + ROWS * OUT_PITCH) * sizeof(float); // ~132 KB < 320 KB WGP LDS

    rcell_kernel<<<dim3(B / ROWS), dim3(ROWS), shmem, stream>>>(
        in, A_, D_, E_, Bm, covin, xs, x0p, out);
}